// SpatialTransformer_1881195676096
// MI455X (gfx1250) — compile-verified
//
#include <hip/hip_runtime.h>
#include <stdint.h>

typedef float v4f __attribute__((ext_vector_type(4)));
typedef int   v4i __attribute__((ext_vector_type(4)));

#define DD   160
#define HHH  192
#define WWW  160
#define HWW  (HHH * WWW)          // 30720
#define VOL  (DD * HHH * WWW)     // 4,915,200 voxels per batch
#define NBATCH 2
#define TOTAL (NBATCH * VOL)      // 9,830,400
#define TPB  256                  // threads per block (8 waves of 32)
#define VPT  4                    // voxels per thread (one float4)
#define VPB  (TPB * VPT)          // 1024 voxels per tile
#define NTILES (TOTAL / VPB)      // 9600 (exact)
#define TILES_PER_BATCH (VOL / VPB) // 4800; 4800 % TPBLK == 0 -> block never straddles batch
#define TPBLK 4                   // tiles per block (double-buffered pipeline)
#define NBLK  (NTILES / TPBLK)    // 2400 blocks

// ---- CDNA5 async global->LDS path (guarded: fall back to NT loads) --------
#if defined(__has_builtin)
#if __has_builtin(__builtin_amdgcn_global_load_async_to_lds_b128)
#define HAVE_ASYNC_LDS 1
#endif
#endif
#ifndef HAVE_ASYNC_LDS
#define HAVE_ASYNC_LDS 0
#endif

#if HAVE_ASYNC_LDS
typedef __attribute__((address_space(1))) v4i gv4i;  // global v4i
typedef __attribute__((address_space(3))) v4i lv4i;  // LDS v4i
#if __has_builtin(__builtin_amdgcn_s_wait_asynccnt)
#define WAIT_ASYNC(n) __builtin_amdgcn_s_wait_asynccnt(n)
#else
#define WAIT_ASYNC(n) asm volatile("s_wait_asynccnt %0" ::"i"(n))
#endif
#endif

static __device__ __forceinline__ int iclamp(int v, int hi) {
  v = v < 0 ? 0 : v;               // min(max(v,0),hi) -> v_med3_i32
  return v > hi ? hi : v;
}

static __device__ __forceinline__ float ffract(float x) {
#if defined(__has_builtin) && __has_builtin(__builtin_amdgcn_fractf)
  return __builtin_amdgcn_fractf(x);     // v_fract_f32 = x - floor(x)
#else
  return x - floorf(x);
#endif
}

// Force a (block-uniform) pointer into SGPRs so vmem selects saddr+voffset.
static __device__ __forceinline__ const float* uniform_base(const float* p) {
  uint64_t u  = (uint64_t)(uintptr_t)p;
  uint32_t lo = __builtin_amdgcn_readfirstlane((uint32_t)u);
  uint32_t hi = __builtin_amdgcn_readfirstlane((uint32_t)(u >> 32));
  return (const float*)(uintptr_t)(((uint64_t)hi << 32) | (uint64_t)lo);
}

__global__ __launch_bounds__(TPB) void warp3d_tri_kernel(
    const float* __restrict__ src,   // [2,1,D,H,W]
    const float* __restrict__ flow,  // [2,3,D,H,W]
    float* __restrict__ out)         // [2,1,D,H,W]
{
  const int tid   = threadIdx.x;
  const int tile0 = blockIdx.x * TPBLK;

  // Block-uniform scalars (whole block stays inside one batch).
  const int nb  = tile0 / TILES_PER_BATCH;
  const int vb0 = tile0 * VPB - nb * VOL;          // voxel base of tile 0
  const float* sp  = src + (size_t)nb * VOL;       // SGPR gather base
  const char*  spb = (const char*)sp;
  const float* fb  = flow + (size_t)(3 * nb) * VOL + vb0;
  float*       ob  = out + (size_t)nb * VOL + vb0;

#if HAVE_ASYNC_LDS
  __shared__ v4f sh[2][3][TPB];  // 24,576 B of the 320 KB WGP LDS
  auto issue = [&](int it, int buf) {
#pragma unroll
    for (int c = 0; c < 3; ++c) {
      const float* fc = uniform_base(fb + (size_t)c * VOL + it * VPB);
      __builtin_amdgcn_global_load_async_to_lds_b128(
          (gv4i*)fc + tid,               // saddr + tid*16B (scale_offset)
          (lv4i*)&sh[buf][c][tid], 0, 0);
    }
  };
  issue(0, 0);
#endif

  // Initial (x, y, z) for this lane's first voxel; updated incrementally:
  // +VPB voxels per tile = 6 rows + 64, with at most one wrap per axis.
  const int v0 = vb0 + tid * VPT;
  int x0 = v0 % WWW;
  int y  = (v0 / WWW) % HHH;
  int z  = v0 / HWW;

#pragma unroll
  for (int i = 0; i < TPBLK; ++i) {
    v4f fx4, fy4, fz4;
#if HAVE_ASYNC_LDS
    if (i + 1 < TPBLK) { issue(i + 1, (i + 1) & 1); WAIT_ASYNC(3); }
    else               { WAIT_ASYNC(0); }
    fx4 = sh[i & 1][0][tid];
    fy4 = sh[i & 1][1][tid];
    fz4 = sh[i & 1][2][tid];
#else
    const float* f0 = uniform_base(fb + i * VPB);
    const float* f1 = uniform_base(fb + (size_t)VOL + i * VPB);
    const float* f2 = uniform_base(fb + (size_t)2 * VOL + i * VPB);
    fx4 = __builtin_nontemporal_load((const v4f*)f0 + tid);
    fy4 = __builtin_nontemporal_load((const v4f*)f1 + tid);
    fz4 = __builtin_nontemporal_load((const v4f*)f2 + tid);
#endif

    v4f res;
#pragma unroll
    for (int j = 0; j < VPT; ++j) {
      // Normalized-grid + flow collapses to: pixel + flow * (dim/2)
      const float ix = (float)(x0 + j) + fx4[j] * (0.5f * WWW);
      const float iy = (float)y        + fy4[j] * (0.5f * HHH);
      const float iz = (float)z        + fz4[j] * (0.5f * DD);
      const float tx = ffract(ix), ty = ffract(iy), tz = ffract(iz);
      const int ixl = (int)floorf(ix);   // -> v_cvt_flr_i32_f32
      const int iyl = (int)floorf(iy);
      const int izl = (int)floorf(iz);
      const int ixh = ixl + 1, iyh = iyl + 1, izh = izl + 1;

      // Zero-padding folded into per-axis weights (6 selects total).
      const float wx0 = ((unsigned)ixl < (unsigned)WWW) ? 1.0f - tx : 0.0f;
      const float wx1 = ((unsigned)ixh < (unsigned)WWW) ? tx        : 0.0f;
      const float wy0 = ((unsigned)iyl < (unsigned)HHH) ? 1.0f - ty : 0.0f;
      const float wy1 = ((unsigned)iyh < (unsigned)HHH) ? ty        : 0.0f;
      const float wz0 = ((unsigned)izl < (unsigned)DD)  ? 1.0f - tz : 0.0f;
      const float wz1 = ((unsigned)izh < (unsigned)DD)  ? tz        : 0.0f;

      // Clamped, byte-scaled offsets (always-legal addresses).
      const int xb0 = iclamp(ixl, WWW - 1) * 4;
      const int xb1 = iclamp(ixh, WWW - 1) * 4;
      const int yb0 = iclamp(iyl, HHH - 1) * (WWW * 4);
      const int yb1 = iclamp(iyh, HHH - 1) * (WWW * 4);
      const int zb0 = iclamp(izl, DD - 1) * (HWW * 4);
      const int zb1 = iclamp(izh, DD - 1) * (HWW * 4);

      const int o00 = zb0 + yb0, o01 = zb0 + yb1;
      const int o10 = zb1 + yb0, o11 = zb1 + yb1;
      const float w00 = wz0 * wy0, w01 = wz0 * wy1;
      const float w10 = wz1 * wy0, w11 = wz1 * wy1;

      float acc;
      acc = *(const float*)(spb + (o00 + xb0)) * (w00 * wx0);
      acc = fmaf(*(const float*)(spb + (o00 + xb1)), w00 * wx1, acc);
      acc = fmaf(*(const float*)(spb + (o01 + xb0)), w01 * wx0, acc);
      acc = fmaf(*(const float*)(spb + (o01 + xb1)), w01 * wx1, acc);
      acc = fmaf(*(const float*)(spb + (o10 + xb0)), w10 * wx0, acc);
      acc = fmaf(*(const float*)(spb + (o10 + xb1)), w10 * wx1, acc);
      acc = fmaf(*(const float*)(spb + (o11 + xb0)), w11 * wx0, acc);
      acc = fmaf(*(const float*)(spb + (o11 + xb1)), w11 * wx1, acc);
      res[j] = acc;
    }
    // Streaming output: keep it out of L2 (reserve L2 for src gathers).
    __builtin_nontemporal_store(res, (v4f*)(ob + i * VPB) + tid);

    // Advance (x0, y, z) by VPB = 6*WWW + 64 voxels (single wrap per axis).
    x0 += 64;
    y  += 6 + (x0 >= WWW ? 1 : 0);
    x0 -= (x0 >= WWW ? WWW : 0);
    z  += (y >= HHH ? 1 : 0);
    y  -= (y >= HHH ? HHH : 0);
  }
}

extern "C" void kernel_launch(void* const* d_in, const int* in_sizes, int n_in,
                              void* d_out, int out_size, void* d_ws, size_t ws_size,
                              hipStream_t stream) {
  (void)in_sizes; (void)n_in; (void)out_size; (void)d_ws; (void)ws_size;
  const float* src  = (const float*)d_in[0];
  const float* flow = (const float*)d_in[1];
  float* out = (float*)d_out;
  warp3d_tri_kernel<<<dim3(NBLK), dim3(TPB), 0, stream>>>(src, flow, out);
}